// DistGConv_6545530159139
// MI455X (gfx1250) — compile-verified
//
#include <hip/hip_runtime.h>
#include <hip/hip_bf16.h>

// Z = segment_sum(val * x[src] -> dst) @ W   reordered as
// Y = X @ W  (dense fp32 WMMA GEMM, Y in d_ws, L2-resident at 51.2 MB)
// Z[dst] += val * Y[src]  (wave-per-edge atomic scatter into d_out)

typedef __attribute__((ext_vector_type(2))) float v2f;
typedef __attribute__((ext_vector_type(8))) float v8f;

#define DFEAT 128

// ---------------------------------------------------------------------------
// Kernel 1: Y = X @ W, fp32, via V_WMMA_F32_16X16X4_F32.
// One wave32 per 16x16 output tile; K-loop 128 in steps of 4.
// ---------------------------------------------------------------------------
__global__ __launch_bounds__(256) void gconv_gemm_xw(const float* __restrict__ X,
                                                     const float* __restrict__ W,
                                                     float* __restrict__ Y,
                                                     int mTiles) {
    const int wave = (int)((blockIdx.x * blockDim.x + threadIdx.x) >> 5);
    const int lane = (int)(threadIdx.x & 31);
    const int nTiles = DFEAT / 16;              // 8
    const int mt = wave / nTiles;
    const int nt = wave % nTiles;
    if (mt >= mTiles) return;

    const int hi  = lane >> 4;                  // 0: lanes 0-15, 1: lanes 16-31
    const int l15 = lane & 15;

    // A fragment: lane -> row (mt*16 + l15); VGPR pair holds K = k+2*hi, k+2*hi+1
    const float* Arow = X + (size_t)(mt * 16 + l15) * DFEAT;
    // B fragment: lane -> col (nt*16 + l15); half-wave K split mirrors A
    const int ncol = nt * 16 + l15;

    v8f acc = {};
#pragma unroll 8
    for (int k = 0; k < DFEAT; k += 4) {
        v2f a, b;
        const int ka = k + 2 * hi;
        a.x = Arow[ka + 0];
        a.y = Arow[ka + 1];
        b.x = W[(size_t)(ka + 0) * DFEAT + ncol];
        b.y = W[(size_t)(ka + 1) * DFEAT + ncol];
        // 8 args: (neg_a, A, neg_b, B, c_mod, C, reuse_a, reuse_b)
        acc = __builtin_amdgcn_wmma_f32_16x16x4_f32(
            false, a, false, b, (short)0, acc, false, false);
    }

    // C/D layout: VGPR i -> row (mt*16 + i + 8*hi), col (nt*16 + l15)
    float* Yt = Y + (size_t)(mt * 16 + 8 * hi) * DFEAT + ncol;
#pragma unroll
    for (int i = 0; i < 8; ++i) {
        Yt[(size_t)i * DFEAT] = acc[i];
    }
}

// ---------------------------------------------------------------------------
// Kernel 2: Z[dst] += val * Y[src].  One wave32 per edge: coalesced 512B
// float4 row gather (L2-resident), 4 global f32 atomics per lane.
// ---------------------------------------------------------------------------
__global__ __launch_bounds__(256) void gconv_scatter(const float* __restrict__ Y,
                                                     const int* __restrict__ esrc,
                                                     const int* __restrict__ edst,
                                                     const float* __restrict__ eval,
                                                     float* __restrict__ Z,
                                                     int nEdges) {
    const int wave = (int)((blockIdx.x * blockDim.x + threadIdx.x) >> 5);
    const int lane = (int)(threadIdx.x & 31);
    if (wave >= nEdges) return;

    const int   s = esrc[wave];   // uniform across wave -> scalarized
    const int   d = edst[wave];
    const float v = eval[wave];

    const float4 y = ((const float4*)(Y + (size_t)s * DFEAT))[lane];
    float* zr = Z + (size_t)d * DFEAT + lane * 4;
    atomicAdd(zr + 0, v * y.x);
    atomicAdd(zr + 1, v * y.y);
    atomicAdd(zr + 2, v * y.z);
    atomicAdd(zr + 3, v * y.w);
}

// ---------------------------------------------------------------------------
extern "C" void kernel_launch(void* const* d_in, const int* in_sizes, int n_in,
                              void* d_out, int out_size, void* d_ws, size_t ws_size,
                              hipStream_t stream) {
    const float* X    = (const float*)d_in[0];   // [N, 128]
    const float* W    = (const float*)d_in[1];   // [128, 128]
    const int*   esrc = (const int*)  d_in[2];   // [E]
    const int*   edst = (const int*)  d_in[3];   // [E]
    const float* eval = (const float*)d_in[4];   // [E]
    float*       Z    = (float*)d_out;           // [N, 128]
    float*       Y    = (float*)d_ws;            // scratch [N, 128] = 51.2 MB

    const int nNodes = in_sizes[0] / DFEAT;
    const int nEdges = in_sizes[2];

    // --- GEMM: Y = X @ W ---
    const int mTiles   = nNodes / 16;            // 100000/16 = 6250, exact
    const int nTiles   = DFEAT / 16;             // 8
    const int nWaves   = mTiles * nTiles;
    const int block    = 256;                    // 8 waves per block
    const int gemmGrid = (nWaves * 32 + block - 1) / block;
    gconv_gemm_xw<<<gemmGrid, block, 0, stream>>>(X, W, Y, mTiles);

    // --- Zero output, then scatter-accumulate edges ---
    hipMemsetAsync(d_out, 0, (size_t)out_size * sizeof(float), stream);

    const int edgeGrid = (nEdges + 7) / 8;       // 8 edges (waves) per block
    gconv_scatter<<<edgeGrid, block, 0, stream>>>(Y, esrc, edst, eval, Z, nEdges);
}